// ControEncodNet_12197707120650
// MI455X (gfx1250) — compile-verified
//
#include <hip/hip_runtime.h>
#include <math.h>

typedef __attribute__((ext_vector_type(2))) float v2f;
typedef __attribute__((ext_vector_type(4))) float v4f;
typedef __attribute__((ext_vector_type(8))) float v8f;
typedef __attribute__((ext_vector_type(4))) int v4i;

#define NROWS 8192
#define INFEATS 128
#define HID1 256
#define HID2 128
#define HID3 64
#define NCLUST 16

static __device__ __forceinline__ v8f wmma_k4(v2f a, v2f b, v8f c) {
  // D = A(16x4 f32) * B(4x16 f32) + C(16x16 f32)
  return __builtin_amdgcn_wmma_f32_16x16x4_f32(false, a, false, b, (short)0, c,
                                               false, false);
}

// Async DMA: 16 bytes per lane, global -> LDS, tracked by ASYNCcnt (no VGPR
// round-trip, no LOADcnt/DScnt). Builtin takes int4* operands (AS1 src, AS3
// dst) per the hipcc diagnostic.
static __device__ __forceinline__ void async_copy16(const float* g, float* l) {
  __builtin_amdgcn_global_load_async_to_lds_b128(
      (__attribute__((address_space(1))) v4i*)(g),
      (__attribute__((address_space(3))) v4i*)(l), 0, 0);
}

// ---------------------------------------------------------------------------
// Kernel 1: T = F_ @ X   [8192x8192]x[8192x128] -> [8192x128]
// block = 256 threads (8 waves). Block owns one 16-row stripe; wave w owns
// columns [16w, 16w+16). F_ (the 268MB HBM stream) is staged into LDS with
// double-buffered async global->LDS DMA so the next chunk's copy overlaps the
// current chunk's 16 WMMAs. X is L2-resident and streamed directly.
// ---------------------------------------------------------------------------
__global__ __launch_bounds__(256) void k_gemm_fx(const float* __restrict__ F,
                                                 const float* __restrict__ X,
                                                 float* __restrict__ T) {
  __shared__ float ldsA[2][16][68];  // 2 x (16 x 64 chunk, +4 pad)
  const int tid  = threadIdx.x;
  const int lane = tid & 31;
  const int wave = tid >> 5;
  const int m  = lane & 15;
  const int hi = lane >> 4;
  const int i0 = blockIdx.x * 16;
  const int c0 = wave * 16;
  const int row = tid >> 4;        // 0..15
  const int col = (tid & 15) * 4;  // 0..60
  const float* gsrc = F + (size_t)(i0 + row) * NROWS + col;

  async_copy16(gsrc, &ldsA[0][row][col]);  // prologue: chunk 0 -> buf 0

  v8f acc = {};
  for (int kc = 0; kc < NROWS; kc += 64) {
    const int buf = (kc >> 6) & 1;
    if (kc + 64 < NROWS) {  // uniform branch
      async_copy16(gsrc + kc + 64, &ldsA[buf ^ 1][row][col]);
      // 2 outstanding; async ops complete in order -> oldest (buf) has landed
      asm volatile("s_wait_asynccnt 0x1" ::: "memory");
    } else {
      asm volatile("s_wait_asynccnt 0x0" ::: "memory");
    }
    __syncthreads();  // all waves' async stripes now visible
#pragma unroll
    for (int s = 0; s < 16; ++s) {
      const int k = 4 * s;
      v2f a, b;
      a.x = ldsA[buf][m][k + 2 * hi];
      a.y = ldsA[buf][m][k + 2 * hi + 1];
      b.x = X[(size_t)(kc + k + 2 * hi) * INFEATS + c0 + m];
      b.y = X[(size_t)(kc + k + 2 * hi + 1) * INFEATS + c0 + m];
      acc = wmma_k4(a, b, acc);
    }
    __syncthreads();  // protect buf before it is async-overwritten next round
  }
#pragma unroll
  for (int r = 0; r < 8; ++r)
    T[(size_t)(i0 + r + 8 * hi) * INFEATS + c0 + m] = acc[r];
}

// ---------------------------------------------------------------------------
// Kernel 2: fused 3-layer MLP per 16-row stripe.
// h1 = tanh(T@W1+b1) [16x256], h2 = tanh(h1@W2+b2) [16x128],
// feats = sigmoid(h2@W3+b3) [16x64].  Activations live in padded LDS.
// ---------------------------------------------------------------------------
__global__ __launch_bounds__(256) void k_mlp(
    const float* __restrict__ T, const float* __restrict__ w1,
    const float* __restrict__ bb1, const float* __restrict__ w2,
    const float* __restrict__ bb2, const float* __restrict__ w3,
    const float* __restrict__ bb3, float* __restrict__ feats) {
  __shared__ float ldsT[16][132];
  __shared__ float ldsH1[16][260];
  __shared__ float ldsH2[16][132];
  const int tid  = threadIdx.x;
  const int lane = tid & 31;
  const int wave = tid >> 5;
  const int m  = lane & 15;
  const int hi = lane >> 4;
  const int i0 = blockIdx.x * 16;

  {  // stage T stripe 16x128
    const int row = tid >> 4;
    const int col = (tid & 15) * 8;
#pragma unroll
    for (int q = 0; q < 8; ++q)
      ldsT[row][col + q] = T[(size_t)(i0 + row) * INFEATS + col + q];
  }
  __syncthreads();

  // layer 1: 256 output cols -> wave handles tiles {wave, wave+8}
#pragma unroll
  for (int t = 0; t < 2; ++t) {
    const int c0 = (wave + 8 * t) * 16;
    v8f acc = {};
#pragma unroll
    for (int s = 0; s < INFEATS / 4; ++s) {
      const int k = 4 * s;
      v2f a, b;
      a.x = ldsT[m][k + 2 * hi];
      a.y = ldsT[m][k + 2 * hi + 1];
      b.x = w1[(size_t)(k + 2 * hi) * HID1 + c0 + m];
      b.y = w1[(size_t)(k + 2 * hi + 1) * HID1 + c0 + m];
      acc = wmma_k4(a, b, acc);
    }
    const float bias = bb1[c0 + m];
#pragma unroll
    for (int r = 0; r < 8; ++r)
      ldsH1[r + 8 * hi][c0 + m] = tanhf(acc[r] + bias);
  }
  __syncthreads();

  // layer 2: 128 output cols -> wave handles tile `wave`
  {
    const int c0 = wave * 16;
    v8f acc = {};
#pragma unroll
    for (int s = 0; s < HID1 / 4; ++s) {
      const int k = 4 * s;
      v2f a, b;
      a.x = ldsH1[m][k + 2 * hi];
      a.y = ldsH1[m][k + 2 * hi + 1];
      b.x = w2[(size_t)(k + 2 * hi) * HID2 + c0 + m];
      b.y = w2[(size_t)(k + 2 * hi + 1) * HID2 + c0 + m];
      acc = wmma_k4(a, b, acc);
    }
    const float bias = bb2[c0 + m];
#pragma unroll
    for (int r = 0; r < 8; ++r)
      ldsH2[r + 8 * hi][c0 + m] = tanhf(acc[r] + bias);
  }
  __syncthreads();

  // layer 3: 64 output cols -> waves 0..3 (wave-uniform branch, EXEC full)
  if (wave < 4) {
    const int c0 = wave * 16;
    v8f acc = {};
#pragma unroll
    for (int s = 0; s < HID2 / 4; ++s) {
      const int k = 4 * s;
      v2f a, b;
      a.x = ldsH2[m][k + 2 * hi];
      a.y = ldsH2[m][k + 2 * hi + 1];
      b.x = w3[(size_t)(k + 2 * hi) * HID3 + c0 + m];
      b.y = w3[(size_t)(k + 2 * hi + 1) * HID3 + c0 + m];
      acc = wmma_k4(a, b, acc);
    }
    const float bias = bb3[c0 + m];
#pragma unroll
    for (int r = 0; r < 8; ++r) {
      const float x = acc[r] + bias;
      feats[(size_t)(i0 + r + 8 * hi) * HID3 + c0 + m] =
          1.0f / (1.0f + expf(-x));
    }
  }
}

// ---------------------------------------------------------------------------
// Kernel 3: sq[i] = sum_k feats[i][k]^2
// ---------------------------------------------------------------------------
__global__ __launch_bounds__(256) void k_sq(const float* __restrict__ feats,
                                            float* __restrict__ sq) {
  const int i = blockIdx.x * 256 + threadIdx.x;
  const v4f* row = (const v4f*)(feats + (size_t)i * HID3);
  float s = 0.0f;
#pragma unroll
  for (int q = 0; q < HID3 / 4; ++q) {
    v4f v = row[q];
    s += v.x * v.x + v.y * v.y + v.z * v.z + v.w * v.w;
  }
  sq[i] = s;
}

// ---------------------------------------------------------------------------
// Kernel 4: fused  out2 = (dist * offdiag * path_forces) @ members
// One wave per 16-row i-tile; loop over 512 j-tiles:
//   prefetch next pf tile -> Gram(16x16) via 16x WMMA (K=64, A in regs, B is
//   the same contiguous pattern by symmetry) -> dist -> mask * pf (streamed
//   non-temporally: each byte read once, keep L2 for feats/members) ->
//   LDS relayout (C->A) -> 4x WMMA against the 16-cluster members tile.
// ---------------------------------------------------------------------------
__global__ __launch_bounds__(256) void k_force(
    const float* __restrict__ feats, const float* __restrict__ sq,
    const float* __restrict__ pf, const float* __restrict__ members,
    float* __restrict__ out2) {
  __shared__ float ldsF[8][16][17];  // per-wave force tile, padded
  const int tid  = threadIdx.x;
  const int lane = tid & 31;
  const int wave = tid >> 5;
  const int m  = lane & 15;
  const int hi = lane >> 4;
  const int i0 = (blockIdx.x * 8 + wave) * 16;

  v2f va[16];  // A operand (feats i-tile), invariant over j loop
#pragma unroll
  for (int s = 0; s < 16; ++s)
    va[s] = *(const v2f*)(feats + (size_t)(i0 + m) * HID3 + 4 * s + 2 * hi);

  float sqi[8];
#pragma unroll
  for (int r = 0; r < 8; ++r) sqi[r] = sq[i0 + r + 8 * hi];

  v8f oacc = {};
  for (int jt = 0; jt < NROWS / 16; ++jt) {
    const int j0 = jt * 16;
    if (jt + 1 < NROWS / 16) {  // uniform: prefetch next pf tile (8 lines)
#pragma unroll
      for (int r = 0; r < 8; ++r)
        __builtin_prefetch(pf + (size_t)(i0 + r + 8 * hi) * NROWS + j0 + 16 + m,
                           0, 1);
    }
    v8f g = {};
#pragma unroll
    for (int s = 0; s < 16; ++s) {
      v2f b = *(const v2f*)(feats + (size_t)(j0 + m) * HID3 + 4 * s + 2 * hi);
      g = wmma_k4(va[s], b, g);
    }
    const float sqj = sq[j0 + m];
#pragma unroll
    for (int r = 0; r < 8; ++r) {
      const int mr = i0 + r + 8 * hi;
      const float pfv =
          __builtin_nontemporal_load(pf + (size_t)mr * NROWS + j0 + m);
      float d2 = sqi[r] + sqj - 2.0f * g[r];
      float dist = sqrtf(fmaxf(d2, 1e-12f));
      if (mr == j0 + m) dist = 0.0f;  // off-diagonal mask
      ldsF[wave][r + 8 * hi][m] = dist * pfv;
    }
    // C-layout -> A-layout relayout through LDS; DS ops of one wave are
    // in-order, but make the store->load data dependency explicit.
    asm volatile("s_wait_dscnt 0" ::: "memory");
#pragma unroll
    for (int s2 = 0; s2 < 4; ++s2) {
      const int k2 = 4 * s2;
      v2f a2, b2;
      a2.x = ldsF[wave][m][k2 + 2 * hi];
      a2.y = ldsF[wave][m][k2 + 2 * hi + 1];
      b2.x = members[(size_t)(j0 + k2 + 2 * hi) * NCLUST + m];
      b2.y = members[(size_t)(j0 + k2 + 2 * hi + 1) * NCLUST + m];
      oacc = wmma_k4(a2, b2, oacc);
    }
  }
#pragma unroll
  for (int r = 0; r < 8; ++r)
    out2[(size_t)(i0 + r + 8 * hi) * NCLUST + m] = oacc[r];
}

// ---------------------------------------------------------------------------
extern "C" void kernel_launch(void* const* d_in, const int* in_sizes, int n_in,
                              void* d_out, int out_size, void* d_ws,
                              size_t ws_size, hipStream_t stream) {
  const float* F   = (const float*)d_in[0];  // F_          [8192,8192]
  const float* X   = (const float*)d_in[1];  // X           [8192,128]
  const float* pf  = (const float*)d_in[2];  // path_forces [8192,8192]
  const float* mem = (const float*)d_in[3];  // members     [8192,16]
  const float* w1  = (const float*)d_in[4];
  const float* b1  = (const float*)d_in[5];
  const float* w2  = (const float*)d_in[6];
  const float* b2  = (const float*)d_in[7];
  const float* w3  = (const float*)d_in[8];
  const float* b3  = (const float*)d_in[9];

  float* feats = (float*)d_out;                 // [8192,64]
  float* out2  = feats + (size_t)NROWS * HID3;  // [8192,16]
  float* T     = (float*)d_ws;                  // [8192,128] scratch
  float* sq    = T + (size_t)NROWS * INFEATS;   // [8192]

  k_gemm_fx<<<NROWS / 16, 256, 0, stream>>>(F, X, T);
  k_mlp<<<NROWS / 16, 256, 0, stream>>>(T, w1, b1, w2, b2, w3, b3, feats);
  k_sq<<<NROWS / 256, 256, 0, stream>>>(feats, sq);
  k_force<<<NROWS / 128, 256, 0, stream>>>(feats, sq, pf, mem, out2);
}